// ChebConv_74148315398480
// MI455X (gfx1250) — compile-verified
//
#include <hip/hip_runtime.h>
#include <hip/hip_bf16.h>
#include <stdint.h>

typedef float v2f __attribute__((ext_vector_type(2)));
typedef float v8f __attribute__((ext_vector_type(8)));

#define DFEAT 96
#define NT    (DFEAT / 16)   // 6 column tiles
#define KSTEPS (DFEAT / 4)   // 24 wmma steps over K

// ---------------------------------------------------------------------------
// Utility: zero a float buffer
// ---------------------------------------------------------------------------
__global__ void zero_kernel(float* __restrict__ p, long long n) {
    long long i = (long long)blockIdx.x * blockDim.x + threadIdx.x;
    if (i < n) p[i] = 0.0f;
}

// ---------------------------------------------------------------------------
// Degree: deg[row[e]] += 1
// ---------------------------------------------------------------------------
__global__ void degree_kernel(const int* __restrict__ rows,
                              float* __restrict__ deg, int nEdges) {
    int e = blockIdx.x * blockDim.x + threadIdx.x;
    if (e < nEdges) atomicAdd(&deg[rows[e]], 1.0f);
}

// dinv[i] = deg>0 ? rsqrt(deg) : 0
__global__ void dinv_kernel(const float* __restrict__ deg,
                            float* __restrict__ dinv, int n) {
    int i = blockIdx.x * blockDim.x + threadIdx.x;
    if (i < n) {
        float d = deg[i];
        dinv[i] = (d > 0.0f) ? rsqrtf(d) : 0.0f;
    }
}

// ---------------------------------------------------------------------------
// prop: out[row] += (-dinv[row]*dinv[col]) * h[col]
// 1 thread per (edge, 4 features): float4 gather + 4 atomic f32 scatter-adds
// ---------------------------------------------------------------------------
__global__ void prop_kernel(const float* __restrict__ h,
                            const int* __restrict__ rows,
                            const int* __restrict__ cols,
                            const float* __restrict__ dinv,
                            float* __restrict__ out, int nEdges) {
    long long t = (long long)blockIdx.x * blockDim.x + threadIdx.x;
    long long e = t / (DFEAT / 4);
    int fc = (int)(t % (DFEAT / 4)) * 4;
    if (e >= nEdges) return;
    int r = rows[e];
    int c = cols[e];
    float w = -dinv[r] * dinv[c];
    const float4 hv = *(const float4*)(h + (size_t)c * DFEAT + fc);
    float* o = out + (size_t)r * DFEAT + fc;
    atomicAdd(o + 0, w * hv.x);
    atomicAdd(o + 1, w * hv.y);
    atomicAdd(o + 2, w * hv.z);
    atomicAdd(o + 3, w * hv.w);
}

// ---------------------------------------------------------------------------
// Chebyshev combine (in place): p = 2*p - q
// ---------------------------------------------------------------------------
__global__ void combine_kernel(float* __restrict__ p,
                               const float* __restrict__ q, long long n) {
    long long i = (long long)blockIdx.x * blockDim.x + threadIdx.x;
    if (i < n) p[i] = 2.0f * p[i] - q[i];
}

// ---------------------------------------------------------------------------
// Dense GEMM with CDNA5 WMMA (f32, full precision): Out (+)= A[N,96] @ W[96,96]
// One wave32 per 16x16 output tile; 24 x v_wmma_f32_16x16x4_f32 over K=96.
// W is staged in LDS (36 KB) once per block, shared by 8 waves.
//
// Software pipelining: ALL A fragments (24 x v2f = 48 VGPRs) and ALL B
// fragments (48 VGPRs) are preloaded back-to-back so the 24 WMMAs issue as
// a pure accumulation chain with no per-iteration memory waits.
//
// ISA fragment layouts (CDNA5 ISA):
//   A 16x4:  lanes 0-15 -> M=lane, VGPR{0,1}=K{0,1}; lanes 16-31 -> K{2,3}
//   B 4x16:  lanes 0-15 -> N=lane, VGPR{0,1}=K{0,1}; lanes 16-31 -> K{2,3}
//   C/D 16x16 f32: VGPR v, lane L -> M = v + 8*(L>=16), N = L%16
// ---------------------------------------------------------------------------
__global__ __launch_bounds__(256)
void gemm_wmma_f32(const float* __restrict__ A,
                   const float* __restrict__ W,
                   const float* __restrict__ bias,
                   float* __restrict__ Out,
                   int nRows, int nTiles, int accumulate) {
    __shared__ float ldsW[DFEAT * DFEAT];  // 36 KB of 320 KB/WGP
    for (int i = threadIdx.x; i < DFEAT * DFEAT; i += blockDim.x)
        ldsW[i] = W[i];
    __syncthreads();

    const int wave = threadIdx.x >> 5;   // wave-uniform
    const int lane = threadIdx.x & 31;
    const int tile = blockIdx.x * 8 + wave;
    if (tile >= nTiles) return;          // whole-wave uniform exit

    const int tileM = tile / NT;
    const int tileN = tile % NT;

    const int hi = lane >> 4;            // 0 | 1
    const int lm = lane & 15;
    const int n  = tileN * 16 + lm;      // B/C/D column for this lane
    int m = tileM * 16 + lm;             // A row for this lane
    if (m >= nRows) m = nRows - 1;       // clamp (N is a multiple of 16 anyway)

    const int rowBase = tileM * 16 + hi * 8;

    // ---- preload the full A row fragment set: 24 contiguous-pair loads ----
    const float* arow = A + (size_t)m * DFEAT + 2 * hi;
    v2f afrag[KSTEPS];
#pragma unroll
    for (int kk = 0; kk < KSTEPS; ++kk) {
        afrag[kk].x = arow[4 * kk];
        afrag[kk].y = arow[4 * kk + 1];
    }

    // ---- preload the full B fragment set from LDS ----
    const float* brow = ldsW + 2 * hi * DFEAT + n;
    v2f bfrag[KSTEPS];
#pragma unroll
    for (int kk = 0; kk < KSTEPS; ++kk) {
        bfrag[kk].x = brow[(4 * kk) * DFEAT];
        bfrag[kk].y = brow[(4 * kk + 1) * DFEAT];
    }

    // ---- accumulator init (bias on first hop, else read-modify-write) ----
    v8f c;
    if (accumulate) {
#pragma unroll
        for (int v = 0; v < 8; ++v) {
            int r = rowBase + v;
            c[v] = Out[(size_t)(r < nRows ? r : nRows - 1) * DFEAT + n];
        }
    } else {
        float bv = bias[n];
#pragma unroll
        for (int v = 0; v < 8; ++v) c[v] = bv;
    }

    // ---- 24 back-to-back WMMAs, no memory waits in the chain ----
#pragma unroll
    for (int kk = 0; kk < KSTEPS; ++kk) {
        // (neg_a, A, neg_b, B, c_mod, C, reuse_a, reuse_b)
        c = __builtin_amdgcn_wmma_f32_16x16x4_f32(false, afrag[kk], false,
                                                  bfrag[kk], (short)0, c,
                                                  false, false);
    }

#pragma unroll
    for (int v = 0; v < 8; ++v) {
        int r = rowBase + v;
        if (r < nRows) Out[(size_t)r * DFEAT + n] = c[v];
    }
}

// ---------------------------------------------------------------------------
// Launcher
// ---------------------------------------------------------------------------
extern "C" void kernel_launch(void* const* d_in, const int* in_sizes, int n_in,
                              void* d_out, int out_size, void* d_ws, size_t ws_size,
                              hipStream_t stream) {
    const float* x    = (const float*)d_in[0];       // [N, 96]
    const int*   ei   = (const int*)d_in[1];         // [2, E]
    const float* w    = (const float*)d_in[2];       // [4, 96, 96]
    const float* bias = (const float*)d_in[3];       // [96]
    float* out = (float*)d_out;

    const int N = in_sizes[0] / DFEAT;
    const int E = in_sizes[1] / 2;
    const int* rows = ei;
    const int* cols = ei + E;
    const size_t wk = (size_t)DFEAT * DFEAT;         // 9216 per hop

    // Workspace layout (aligned): deg[N], dinv[N], bufA/B/C [N*96]
    char* ws = (char*)d_ws;
    float* deg  = (float*)ws;
    float* dinv = deg + N;
    size_t small = (((size_t)2 * N * sizeof(float)) + 255) & ~(size_t)255;
    const long long feat = (long long)N * DFEAT;
    float* bufA = (float*)(ws + small);
    float* bufB = bufA + feat;
    float* bufC = bufB + feat;

    const int B256 = 256;
    const int gN    = (N + B256 - 1) / B256;
    const int gE    = (E + B256 - 1) / B256;
    const int gF    = (int)((feat + B256 - 1) / B256);
    const long long propT = (long long)E * (DFEAT / 4);
    const int gP    = (int)((propT + B256 - 1) / B256);
    const int mTiles = (N + 15) / 16;
    const int nTiles = mTiles * NT;
    const int gG    = (nTiles + 7) / 8;

    // ---- normalization ----
    zero_kernel<<<gN, B256, 0, stream>>>(deg, N);
    degree_kernel<<<gE, B256, 0, stream>>>(rows, deg, E);
    dinv_kernel<<<gN, B256, 0, stream>>>(deg, dinv, N);

    // ---- k = 0: out = x @ W0 + bias ----
    gemm_wmma_f32<<<gG, B256, 0, stream>>>(x, w + 0 * wk, bias, out, N, nTiles, 0);

    // ---- k = 1: tx1 = L_hat x ; out += tx1 @ W1 ----
    zero_kernel<<<gF, B256, 0, stream>>>(bufA, feat);
    prop_kernel<<<gP, B256, 0, stream>>>(x, rows, cols, dinv, bufA, E);
    gemm_wmma_f32<<<gG, B256, 0, stream>>>(bufA, w + 1 * wk, bias, out, N, nTiles, 1);

    // ---- k = 2: tx2 = 2 L_hat tx1 - x ; out += tx2 @ W2 ----
    zero_kernel<<<gF, B256, 0, stream>>>(bufB, feat);
    prop_kernel<<<gP, B256, 0, stream>>>(bufA, rows, cols, dinv, bufB, E);
    combine_kernel<<<gF, B256, 0, stream>>>(bufB, x, feat);
    gemm_wmma_f32<<<gG, B256, 0, stream>>>(bufB, w + 2 * wk, bias, out, N, nTiles, 1);

    // ---- k = 3: tx3 = 2 L_hat tx2 - tx1 ; out += tx3 @ W3 ----
    zero_kernel<<<gF, B256, 0, stream>>>(bufC, feat);
    prop_kernel<<<gP, B256, 0, stream>>>(bufB, rows, cols, dinv, bufC, E);
    combine_kernel<<<gF, B256, 0, stream>>>(bufC, bufA, feat);
    gemm_wmma_f32<<<gG, B256, 0, stream>>>(bufC, w + 3 * wk, bias, out, N, nTiles, 1);
}